// GCN_42872363549056
// MI455X (gfx1250) — compile-verified
//
#include <hip/hip_runtime.h>
#include <hip/hip_bf16.h>

typedef __attribute__((ext_vector_type(2))) float v2f;
typedef __attribute__((ext_vector_type(8))) float v8f;
typedef __attribute__((ext_vector_type(4))) unsigned int u32x4;
typedef __attribute__((ext_vector_type(4))) int i32x4;
typedef __attribute__((ext_vector_type(8))) int i32x8;

#define GCN_N 100000
#define GCN_E 640000
#define GCN_D 128

#if __has_builtin(__builtin_amdgcn_tensor_load_to_lds) && \
    __has_builtin(__builtin_amdgcn_s_wait_tensorcnt)
#define GCN_USE_TDM 1
#else
#define GCN_USE_TDM 0
#endif

// ---------------------------------------------------------------------------
// H[nRows x 128] = act(X) @ W    via V_WMMA_F32_16X16X4_F32 (exact fp32 path)
// Block = 256 threads = 8 waves; each wave computes one 16-row strip x 128 cols.
// W (128x128 f32 = 64KB) staged in LDS once per workgroup — by the Tensor Data
// Mover (one 2D-tile DMA, tracked with TENSORcnt) when the builtin is present.
// A layout (ISA 16x4 f32): lane%16 = M, lane/16 selects K-pair {0,1} vs {2,3},
//   VGPR0 = first K of pair, VGPR1 = second.  B mirrored with N across lanes.
// C/D layout: VGPR v, lane L -> M = v + 8*(L/16), N = L%16.
// ---------------------------------------------------------------------------
__global__ __launch_bounds__(256)
void gcn_gemm_wmma(const float* __restrict__ X, const float* __restrict__ W,
                   float* __restrict__ H, int nRows, int reluInput)
{
    __shared__ float Wlds[GCN_D * GCN_D];   // 64 KB of the 320 KB WGP LDS

#if GCN_USE_TDM
    // --- Tensor Data Mover: DMA the whole 128x128 f32 tile into LDS. ---
    if ((threadIdx.x >> 5) == 0) {          // one wave issues; EXEC ignored by TDM
        unsigned long long gaddr = (unsigned long long)(const void*)W;
        unsigned lds_off =
            (unsigned)(unsigned long long)(__attribute__((address_space(3))) float*)Wlds;

        // D# group 0: count=1 | lds_addr | global_addr[56:0] | type=2
        u32x4 g0;
        g0.x = 1u;                                    // count=1, user mode
        g0.y = lds_off;                               // lds_addr (bytes)
        g0.z = (unsigned)(gaddr & 0xFFFFFFFFull);     // global_addr[31:0] -> bits 95:64
        g0.w = (unsigned)((gaddr >> 32) & 0x01FFFFFFull) | (2u << 30); // addr[56:32] | type=2

        // D# group 1: data_size=4B, tensor 128x128, tile 128x128, stride 128
        i32x8 g1;
        g1[0] = (int)(2u << 16);                      // wg_mask=0 | data_size=2 (4B)
        g1[1] = (int)(128u << 16);                    // tensor_dim0 low16 @ bits 63:48
        g1[2] = (int)(128u << 16);                    // dim0 hi16=0 | tensor_dim1 low16
        g1[3] = (int)(128u << 16);                    // dim1 hi16=0 | tile_dim0=128
        g1[4] = (int)128u;                            // tile_dim1=128 | tile_dim2=0
        g1[5] = (int)128u;                            // tensor_dim0_stride low32
        g1[6] = 0;                                    // stride hi16 | dim1_stride lo16
        g1[7] = 0;

        i32x4 z4 = {0, 0, 0, 0};                      // groups 2/3 unused (2D tensor)
#if __clang_major__ >= 23
        i32x8 z8 = {0, 0, 0, 0, 0, 0, 0, 0};
        __builtin_amdgcn_tensor_load_to_lds(g0, g1, z4, z4, z8, 0);
#else
        __builtin_amdgcn_tensor_load_to_lds(g0, g1, z4, z4, 0);
#endif
        __builtin_amdgcn_s_wait_tensorcnt(0);
    }
#else
    // --- Fallback: cooperative float4 staging. ---
    {
        const float4* wg = reinterpret_cast<const float4*>(W);
        float4* ws = reinterpret_cast<float4*>(Wlds);
        for (int i = threadIdx.x; i < (GCN_D * GCN_D) / 4; i += 256)
            ws[i] = wg[i];
    }
#endif
    __syncthreads();

    const int wave  = threadIdx.x >> 5;
    const int lane  = threadIdx.x & 31;
    const int strip = blockIdx.x * 8 + wave;
    const int nStrips = nRows >> 4;
    if (strip >= nStrips) return;           // wave-uniform: EXEC stays all-1s

    const int rowBase = strip << 4;
    const int m     = lane & 15;            // M (for A) / N-within-tile (for B/C)
    const int khalf = lane >> 4;            // 0 -> K{0,1}, 1 -> K{2,3}
    const float* __restrict__ xrow = X + (size_t)(rowBase + m) * GCN_D;

    v8f acc[8] = {};                        // 8 N-tiles of 16x16 f32

    #pragma unroll 2
    for (int k = 0; k < GCN_D; k += 4) {
        const int ka = k + 2 * khalf;
        // Prefetch upcoming A data (speculative; dropped if OOB).
        __builtin_prefetch(xrow + ka + 32, 0, 1);

        float a0 = xrow[ka];
        float a1 = xrow[ka + 1];
        if (reluInput) { a0 = fmaxf(a0, 0.0f); a1 = fmaxf(a1, 0.0f); }
        v2f a; a.x = a0; a.y = a1;

        #pragma unroll
        for (int t = 0; t < 8; ++t) {
            v2f b;
            b.x = Wlds[(ka + 0) * GCN_D + t * 16 + m];
            b.y = Wlds[(ka + 1) * GCN_D + t * 16 + m];
            acc[t] = __builtin_amdgcn_wmma_f32_16x16x4_f32(
                /*neg_a=*/false, a, /*neg_b=*/false, b,
                /*c_mod=*/(short)0, acc[t], /*reuse_a=*/false, /*reuse_b=*/false);
        }
    }

    #pragma unroll
    for (int t = 0; t < 8; ++t) {
        #pragma unroll
        for (int v = 0; v < 8; ++v) {
            const int mm = v + 8 * khalf;
            H[(size_t)(rowBase + mm) * GCN_D + t * 16 + m] = acc[t][v];
        }
    }
}

// ---------------------------------------------------------------------------
// Degree / normalization / bias-init / scatter kernels (memory-bound phase)
// ---------------------------------------------------------------------------
__global__ void gcn_zero_f32(float* __restrict__ p, int n)
{
    int i = blockIdx.x * blockDim.x + threadIdx.x;
    if (i < n) p[i] = 0.0f;
}

__global__ void gcn_degree(const int* __restrict__ col, float* __restrict__ deg,
                           int nE, int nN)
{
    int i = blockIdx.x * blockDim.x + threadIdx.x;
    int total = nE + nN;
    if (i >= total) return;
    int c = (i < nE) ? col[i] : (i - nE);   // implicit self-loops
    atomicAdd(&deg[c], 1.0f);
}

__global__ void gcn_deg_inv_sqrt(float* __restrict__ deg, int n)
{
    int i = blockIdx.x * blockDim.x + threadIdx.x;
    if (i >= n) return;
    float d = deg[i];
    deg[i] = (d > 0.0f) ? rsqrtf(fmaxf(d, 1.0f)) : 0.0f;
}

// out = segment_sum(...) + b  ->  seed the accumulator with the bias.
__global__ void gcn_init_bias(float* __restrict__ out, const float* __restrict__ b,
                              int nRows)
{
    int i = blockIdx.x * blockDim.x + threadIdx.x;
    if (i < nRows * GCN_D) out[i] = b[i & (GCN_D - 1)];
}

// One thread per (edge, 4-feature group): float4 gather of h[row], scale by
// dis[row]*dis[col], 4x global_atomic_add_f32 into acc[col].
__global__ __launch_bounds__(256)
void gcn_scatter(const float* __restrict__ h,
                 const int* __restrict__ row, const int* __restrict__ col,
                 const float* __restrict__ dis, float* __restrict__ acc,
                 int nE, int nN)
{
    long long tid = (long long)blockIdx.x * blockDim.x + threadIdx.x;
    int e = (int)(tid >> 5);
    int g = (int)(tid & 31);                // 4-float feature group 0..31
    int total = nE + nN;
    if (e >= total) return;

    int r, c;
    if (e < nE) { r = row[e]; c = col[e]; }
    else        { r = e - nE; c = r; }      // self-loop

    float norm = dis[r] * dis[c];
    float4 hv = *reinterpret_cast<const float4*>(h + (size_t)r * GCN_D + g * 4);
    float* dst = acc + (size_t)c * GCN_D + g * 4;
    atomicAdd(dst + 0, hv.x * norm);
    atomicAdd(dst + 1, hv.y * norm);
    atomicAdd(dst + 2, hv.z * norm);
    atomicAdd(dst + 3, hv.w * norm);
}

// ---------------------------------------------------------------------------
extern "C" void kernel_launch(void* const* d_in, const int* in_sizes, int n_in,
                              void* d_out, int out_size, void* d_ws, size_t ws_size,
                              hipStream_t stream)
{
    const float* x  = (const float*)d_in[0];          // [N, 128]
    const int*   ei = (const int*)  d_in[1];          // [2, E]
    const float* W1 = (const float*)d_in[2];          // [128, 128]
    const float* b1 = (const float*)d_in[3];          // [128]
    const float* W2 = (const float*)d_in[4];          // [128, 128]
    const float* b2 = (const float*)d_in[5];          // [128]
    float* out = (float*)d_out;                       // [N, 128]

    const int* row = ei;                              // edge_index[0]
    const int* col = ei + GCN_E;                      // edge_index[1]

    // Workspace layout (floats): deg/dis | h | acc1   (~103 MB total)
    float* ws   = (float*)d_ws;
    float* deg  = ws;                                  // N floats (becomes dis)
    float* h    = ws + (1 << 17);                      // 512KB-aligned offset
    float* acc1 = h + (size_t)GCN_N * GCN_D;

    const int totalE = GCN_E + GCN_N;
    const int T = 256;

    // --- normalization: deg -> deg_inv_sqrt ---
    gcn_zero_f32<<<(GCN_N + T - 1) / T, T, 0, stream>>>(deg, GCN_N);
    gcn_degree<<<(totalE + T - 1) / T, T, 0, stream>>>(col, deg, GCN_E, GCN_N);
    gcn_deg_inv_sqrt<<<(GCN_N + T - 1) / T, T, 0, stream>>>(deg, GCN_N);

    const int nStrips    = GCN_N / 16;                 // 6250 (exact)
    const int gemmBlocks = (nStrips + 7) / 8;          // 782
    const long long scatThreads = (long long)totalE * 32;
    const int scatBlocks = (int)((scatThreads + T - 1) / T);
    const int initBlocks = (GCN_N * GCN_D + T - 1) / T;

    // --- layer 1: h1 = x @ W1 ; acc1 = b1 + scatter(h1) ---
    gcn_gemm_wmma<<<gemmBlocks, T, 0, stream>>>(x, W1, h, GCN_N, 0);
    gcn_init_bias<<<initBlocks, T, 0, stream>>>(acc1, b1, GCN_N);
    gcn_scatter<<<scatBlocks, T, 0, stream>>>(h, row, col, deg, acc1, GCN_E, GCN_N);

    // --- layer 2: h2 = relu(acc1) @ W2 ; out = b2 + scatter(h2) ---
    gcn_gemm_wmma<<<gemmBlocks, T, 0, stream>>>(acc1, W2, h, GCN_N, 1);
    gcn_init_bias<<<initBlocks, T, 0, stream>>>(out, b2, GCN_N);
    gcn_scatter<<<scatBlocks, T, 0, stream>>>(h, row, col, deg, out, GCN_E, GCN_N);
}